// WassersteinCov_50053548868122
// MI455X (gfx1250) — compile-verified
//
#include <hip/hip_runtime.h>
#include <hip/hip_bf16.h>
#include <math.h>

typedef __attribute__((ext_vector_type(2))) float v2f;
typedef __attribute__((ext_vector_type(8))) float v8f;

#define NROWS 4096
#define NDIM  2048
#define BM 128
#define BN 64
#define BK 16
#define NS_ITERS 16

// -------------------- column mean / inv-std (ddof=1) --------------------
__global__ void colstats_kernel(const float* __restrict__ z,
                                float* __restrict__ mean,
                                float* __restrict__ invstd,
                                int N, int D) {
    int j = blockIdx.x * blockDim.x + threadIdx.x;
    if (j >= D) return;
    float s = 0.f, ss = 0.f;
    for (int i = 0; i < N; ++i) {
        float v = z[(size_t)i * D + j];
        s += v; ss += v * v;
    }
    float m   = s / (float)N;
    float var = (ss - (float)N * m * m) / (float)(N - 1);
    mean[j]   = m;
    invstd[j] = rsqrtf(var);
}

__global__ void standardize_kernel(const float* __restrict__ z,
                                   const float* __restrict__ mean,
                                   const float* __restrict__ invstd,
                                   float* __restrict__ zs,
                                   size_t total, int D) {
    size_t i = (size_t)blockIdx.x * blockDim.x + threadIdx.x;
    if (i >= total) return;
    int j = (int)(i % D);
    zs[i] = (z[i] - mean[j]) * invstd[j];
}

// -------------------- f32 WMMA GEMM: C = beta*Cin + alpha*op(A)@B (+eps on diag) ---
// op(A) = A^T when transA (A stored [K,M] row-major), else A ([M,K] row-major).
// B stored [K,N] row-major. All dims assumed multiples of tile sizes.
__global__ __launch_bounds__(256) void gemm_f32_wmma(
    const float* __restrict__ A, int lda, int transA,
    const float* __restrict__ B, int ldb,
    const float* __restrict__ Cin,
    float* __restrict__ Cout, int ldc,
    int M, int N, int K,
    float alpha, float beta, float eps_diag)
{
    __shared__ float As[BK][BM + 4];
    __shared__ float Bs[BK][BN + 4];

    const int tid  = threadIdx.x;
    const int lane = tid & 31;
    const int wave = tid >> 5;      // 0..7
    const int wr   = wave & 3;      // wave row 0..3  -> 32-row strips
    const int wc   = wave >> 2;     // wave col 0..1  -> 32-col strips
    const int half = lane >> 4;     // 0: lanes 0-15, 1: lanes 16-31
    const int l16  = lane & 15;

    const int m0 = blockIdx.x * BM;
    const int n0 = blockIdx.y * BN;

    v8f acc[2][2] = {};

    for (int k0 = 0; k0 < K; k0 += BK) {
        // ---- stage A tile into LDS: As[kk][mm] = op(A)[m0+mm][k0+kk]
        if (transA) {
            for (int e = tid; e < BK * BM; e += 256) {
                int kk = e / BM, mm = e % BM;       // contiguous in m -> coalesced
                As[kk][mm] = A[(size_t)(k0 + kk) * lda + (m0 + mm)];
            }
        } else {
            for (int e = tid; e < BK * BM; e += 256) {
                int kk = e % BK, mm = e / BK;       // contiguous in k (16 floats)
                As[kk][mm] = A[(size_t)(m0 + mm) * lda + (k0 + kk)];
            }
        }
        // ---- stage B tile: Bs[kk][nn] = B[k0+kk][n0+nn]
        for (int e = tid; e < BK * BN; e += 256) {
            int kk = e / BN, nn = e % BN;
            Bs[kk][nn] = B[(size_t)(k0 + kk) * ldb + (n0 + nn)];
        }
        // hint the next K tile toward the caches (global_prefetch_b8)
        if (k0 + BK < K) {
            if (transA) __builtin_prefetch(&A[(size_t)(k0 + BK) * lda + m0 + tid], 0, 3);
            else        __builtin_prefetch(&A[(size_t)(m0 + (tid >> 1)) * lda + k0 + BK], 0, 3);
            __builtin_prefetch(&B[(size_t)(k0 + BK) * ldb + n0 + (tid & 63)], 0, 3);
        }
        __syncthreads();

        // ---- 4 WMMA K-steps of 4
        #pragma unroll
        for (int kk = 0; kk < BK; kk += 4) {
            const int ka = kk + half * 2;   // lanes 16-31 take K+2,K+3
            v2f af[2], bf[2];
            #pragma unroll
            for (int si = 0; si < 2; ++si) {
                int m = wr * 32 + si * 16 + l16;
                af[si].x = As[ka][m];
                af[si].y = As[ka + 1][m];
            }
            #pragma unroll
            for (int sj = 0; sj < 2; ++sj) {
                int n = wc * 32 + sj * 16 + l16;
                bf[sj].x = Bs[ka][n];
                bf[sj].y = Bs[ka + 1][n];
            }
            #pragma unroll
            for (int si = 0; si < 2; ++si)
                #pragma unroll
                for (int sj = 0; sj < 2; ++sj)
                    acc[si][sj] = __builtin_amdgcn_wmma_f32_16x16x4_f32(
                        false, af[si], false, bf[sj],
                        (short)0, acc[si][sj], false, false);
        }
        __syncthreads();
    }

    // ---- epilogue: C/D layout = 8 VGPRs; lanes 16-31 hold rows M+8
    #pragma unroll
    for (int si = 0; si < 2; ++si) {
        #pragma unroll
        for (int sj = 0; sj < 2; ++sj) {
            const int rbase = m0 + wr * 32 + si * 16 + half * 8;
            const int col   = n0 + wc * 32 + sj * 16 + l16;
            #pragma unroll
            for (int r = 0; r < 8; ++r) {
                const int row = rbase + r;
                float v = alpha * acc[si][sj][r];
                if (beta != 0.0f) v += beta * Cin[(size_t)row * ldc + col];
                if (eps_diag != 0.0f && row == col) v += eps_diag;
                Cout[(size_t)row * ldc + col] = v;
            }
        }
    }
}

// -------------------- deterministic reductions --------------------
__global__ void trace_diag_kernel(const float* __restrict__ mat, int D, float* slot) {
    __shared__ float red[256];
    float s = 0.f;
    for (int i = threadIdx.x; i < D; i += 256) s += mat[(size_t)i * D + i];
    red[threadIdx.x] = s; __syncthreads();
    for (int w = 128; w > 0; w >>= 1) {
        if (threadIdx.x < w) red[threadIdx.x] += red[threadIdx.x + w];
        __syncthreads();
    }
    if (threadIdx.x == 0) slot[0] = red[0];
}

__global__ void frob_stage1(const float* __restrict__ m, size_t count,
                            float* __restrict__ partials) {
    __shared__ float red[256];
    float s = 0.f;
    for (size_t i = (size_t)blockIdx.x * 256 + threadIdx.x; i < count; i += 256 * 256) {
        float v = m[i]; s += v * v;
    }
    red[threadIdx.x] = s; __syncthreads();
    for (int w = 128; w > 0; w >>= 1) {
        if (threadIdx.x < w) red[threadIdx.x] += red[threadIdx.x + w];
        __syncthreads();
    }
    if (threadIdx.x == 0) partials[blockIdx.x] = red[0];
}

__global__ void frob_stage2(const float* __restrict__ partials, float* slot) {
    __shared__ float red[256];
    red[threadIdx.x] = partials[threadIdx.x]; __syncthreads();
    for (int w = 128; w > 0; w >>= 1) {
        if (threadIdx.x < w) red[threadIdx.x] += red[threadIdx.x + w];
        __syncthreads();
    }
    if (threadIdx.x == 0) slot[0] = sqrtf(red[0]);   // ||.||_F
}

// -------------------- Newton-Schulz helpers --------------------
__global__ void ns_init_kernel(const float* __restrict__ src, const float* __restrict__ c_slot,
                               float* __restrict__ Y, float* __restrict__ Z, int D) {
    size_t i = (size_t)blockIdx.x * 256 + threadIdx.x;
    if (i >= (size_t)D * D) return;
    float c = c_slot[0];
    Y[i] = src[i] / c;
    int row = (int)(i / D), col = (int)(i % D);
    Z[i] = (row == col) ? 1.0f : 0.0f;
}

__global__ void scale_sqrtc_kernel(float* __restrict__ m, const float* __restrict__ c_slot,
                                   size_t count) {
    size_t i = (size_t)blockIdx.x * 256 + threadIdx.x;
    if (i >= count) return;
    m[i] *= sqrtf(c_slot[0]);
}

// slots: 0 trA, 1 trB, 2 ||B||_F, 3 ||M||_F, 4 tr(Y_M)
__global__ void finalize_kernel(const float* __restrict__ slots, float* __restrict__ out) {
    if (threadIdx.x == 0)
        out[0] = slots[0] + slots[1] - 2.0f * sqrtf(slots[3]) * slots[4];
}

// -------------------- host orchestration --------------------
static inline void run_gemm(const float* A, int lda, int transA,
                            const float* B, int ldb,
                            const float* Cin, float* Cout, int ldc,
                            int M, int N, int K,
                            float alpha, float beta, float eps, hipStream_t s) {
    dim3 grid(M / BM, N / BN);
    gemm_f32_wmma<<<grid, 256, 0, s>>>(A, lda, transA, B, ldb, Cin, Cout, ldc,
                                       M, N, K, alpha, beta, eps);
}

extern "C" void kernel_launch(void* const* d_in, const int* in_sizes, int n_in,
                              void* d_out, int out_size, void* d_ws, size_t ws_size,
                              hipStream_t stream) {
    (void)in_sizes; (void)n_in; (void)out_size; (void)ws_size;
    const int N = NROWS, D = NDIM;
    const float* za = (const float*)d_in[0];
    const float* zb = (const float*)d_in[1];
    float* out = (float*)d_out;
    float* ws  = (float*)d_ws;

    // workspace layout (floats)
    float* slots    = ws;              // [0..15]
    float* partials = ws + 16;         // [16..271]
    float* mean_a   = ws + 512;
    float* istd_a   = mean_a + D;
    float* mean_b   = istd_a + D;
    float* istd_b   = mean_b + D;
    float* zsa      = ws + 512 + 4 * (size_t)D;            // N*D
    float* zsb      = zsa + (size_t)N * D;                 // N*D
    float* Amat     = zsb + (size_t)N * D;                 // D*D
    float* Bmat     = Amat + (size_t)D * D;                // D*D
    // NS scratch aliases the (dead after SYRK) standardized matrices:
    float* M0 = zsa;
    float* M1 = zsa + (size_t)D * D;
    float* M2 = zsb;
    float* M3 = zsb + (size_t)D * D;

    const size_t DD = (size_t)D * D;
    const int nb_elem = (int)(((size_t)N * D) / 256);
    const int nb_dd   = (int)(DD / 256);

    // 1) column stats + standardize
    colstats_kernel<<<D / 256, 256, 0, stream>>>(za, mean_a, istd_a, N, D);
    colstats_kernel<<<D / 256, 256, 0, stream>>>(zb, mean_b, istd_b, N, D);
    standardize_kernel<<<nb_elem, 256, 0, stream>>>(za, mean_a, istd_a, zsa, (size_t)N * D, D);
    standardize_kernel<<<nb_elem, 256, 0, stream>>>(zb, mean_b, istd_b, zsb, (size_t)N * D, D);

    // 2) SYRK: A = zsa^T zsa + eps I ; B = zsb^T zsb + eps I  (K = N rows)
    run_gemm(zsa, D, 1, zsa, D, nullptr, Amat, D, D, D, N, 1.0f, 0.0f, 1e-3f, stream);
    run_gemm(zsb, D, 1, zsb, D, nullptr, Bmat, D, D, D, N, 1.0f, 0.0f, 1e-3f, stream);

    // 3) traces (before B is clobbered)
    trace_diag_kernel<<<1, 256, 0, stream>>>(Amat, D, slots + 0);
    trace_diag_kernel<<<1, 256, 0, stream>>>(Bmat, D, slots + 1);

    // 4) sqrtm(B) via coupled Newton-Schulz. c1 = ||B||_F
    frob_stage1<<<256, 256, 0, stream>>>(Bmat, DD, partials);
    frob_stage2<<<1, 256, 0, stream>>>(partials, slots + 2);
    float *Y = M0, *Z = M1, *T = M2, *W = M3;
    ns_init_kernel<<<nb_dd, 256, 0, stream>>>(Bmat, slots + 2, Y, Z, D);
    for (int it = 0; it < NS_ITERS; ++it) {
        run_gemm(Z, D, 0, Y, D, nullptr, T, D, D, D, D, 1.0f, 0.0f, 0.0f, stream);   // T = Z@Y
        run_gemm(Y, D, 0, T, D, Y, W, D, D, D, D, -0.5f, 1.5f, 0.0f, stream);        // W = 1.5Y - 0.5 Y@T
        run_gemm(T, D, 0, Z, D, Z, Y, D, D, D, D, -0.5f, 1.5f, 0.0f, stream);        // Yold <- 1.5Z - 0.5 T@Z
        float* oy = Y; float* oz = Z;
        Y = W; Z = oy; W = oz;                                                        // rotate buffers
    }
    // B12 = sqrt(c1) * Y
    scale_sqrtc_kernel<<<nb_dd, 256, 0, stream>>>(Y, slots + 2, DD);

    // 5) M = B12 @ A @ B12   (G into Z buffer, M into Bmat — B dead)
    run_gemm(Y, D, 0, Amat, D, nullptr, Z, D, D, D, D, 1.0f, 0.0f, 0.0f, stream);    // G = B12@A
    run_gemm(Z, D, 0, Y, D, nullptr, Bmat, D, D, D, D, 1.0f, 0.0f, 0.0f, stream);    // M = G@B12

    // 6) sqrtm(M): only its trace is needed. c2 = ||M||_F
    frob_stage1<<<256, 256, 0, stream>>>(Bmat, DD, partials);
    frob_stage2<<<1, 256, 0, stream>>>(partials, slots + 3);
    Y = M0; Z = M1; T = M2; W = M3;
    ns_init_kernel<<<nb_dd, 256, 0, stream>>>(Bmat, slots + 3, Y, Z, D);
    for (int it = 0; it < NS_ITERS; ++it) {
        run_gemm(Z, D, 0, Y, D, nullptr, T, D, D, D, D, 1.0f, 0.0f, 0.0f, stream);
        run_gemm(Y, D, 0, T, D, Y, W, D, D, D, D, -0.5f, 1.5f, 0.0f, stream);
        run_gemm(T, D, 0, Z, D, Z, Y, D, D, D, D, -0.5f, 1.5f, 0.0f, stream);
        float* oy = Y; float* oz = Z;
        Y = W; Z = oy; W = oz;
    }
    trace_diag_kernel<<<1, 256, 0, stream>>>(Y, D, slots + 4);   // tr(Y) ; tr(C)=sqrt(c2)*tr(Y)

    // 7) out = trA + trB - 2*tr(C)
    finalize_kernel<<<1, 32, 0, stream>>>(slots, out);
}